// JointContrastiveClassificationLoss_60739427500227
// MI455X (gfx1250) — compile-verified
//
#include <hip/hip_runtime.h>
#include <hip/hip_bf16.h>
#include <math.h>

typedef __attribute__((ext_vector_type(2))) float v2f;
typedef __attribute__((ext_vector_type(8))) float v8f;

#define NROWS 2048
#define NCLS  28
#define NDIM  256
#define INV_TEMP 14.285714285714286f   // 1/0.07
#define NTILES (NROWS / 16)            // 128
#define WAVES 4

// ---------------------------------------------------------------------------
// Kernel 1: per-row preprocessing.
//   ew[i]   = clamp(1 - mean_active_entropy/ln2, 0, 1)
//   msk[i]  = 28-bit label bitmask
//   vtab[i][a] = max_{b in set_i} va[a][b]
// Also zeroes the global accumulators.
// ---------------------------------------------------------------------------
__global__ void prep_kernel(const float* __restrict__ logits,
                            const float* __restrict__ labels,
                            const float* __restrict__ va,
                            float* __restrict__ ew,
                            unsigned* __restrict__ msk,
                            float* __restrict__ vtab,
                            float* __restrict__ accum)
{
    int t = blockIdx.x * blockDim.x + threadIdx.x;
    if (t == 0) { accum[0] = 0.0f; accum[1] = 0.0f; }
    if (t >= NROWS) return;

    unsigned m = 0u;
    float labsum = 0.0f, entsum = 0.0f;
    for (int c = 0; c < NCLS; ++c) {
        float lab = labels[t * NCLS + c];
        float lg  = logits[t * NCLS + c];
        float p   = 1.0f / (1.0f + __expf(-lg));
        float ent = -(p * __logf(p + 1e-8f) + (1.0f - p) * __logf(1.0f - p + 1e-8f));
        if (lab > 0.0f) {
            m |= (1u << c);
            labsum += lab;
            entsum += ent * lab;
        }
    }
    float safe = entsum / fmaxf(labsum, 1.0f);
    float w = 1.0f - safe / 0.6931471805599453f;   // ln(2)
    ew[t]  = fminf(fmaxf(w, 0.0f), 1.0f);
    msk[t] = m;

    for (int a = 0; a < NCLS; ++a) {
        float mx = -INFINITY;
        unsigned mm = m;
        while (mm) {
            int b = __ffs(mm) - 1;
            mm &= (mm - 1);
            mx = fmaxf(mx, va[a * NCLS + b]);
        }
        vtab[t * NCLS + a] = mx;
    }
}

// ---------------------------------------------------------------------------
// f32 WMMA 16x16 tile of sim = A-rows dot B-rows over K=256.
// A fragment: lane L -> row L%16, K = kstep + 2*(L/16) + {0,1}
// B fragment: lane L -> col L%16, same K addressing (B = E[j-rows]^T)
// ---------------------------------------------------------------------------
__device__ __forceinline__ v8f sim_tile(const float* __restrict__ Arow,
                                        const float* __restrict__ Brow,
                                        int half)
{
    v8f acc = {};
#pragma unroll 8
    for (int k = 0; k < NDIM; k += 4) {
        const int kk = k + 2 * half;
        v2f a = *(const v2f*)(Arow + kk);
        v2f b = *(const v2f*)(Brow + kk);
        acc = __builtin_amdgcn_wmma_f32_16x16x4_f32(
                  false, a, false, b, (short)0, acc, false, false);
    }
    return acc;
}

// ---------------------------------------------------------------------------
// Kernel 2: fused contrastive loss over one 16-row anchor tile per block.
// Block = 128 threads = 4 waves (wave32); waves split the 128 column tiles.
// Pass 1: row max over j != i.  Pass 2: exp(sim - c) pos/neg accumulation.
// ---------------------------------------------------------------------------
__global__ void __launch_bounds__(128)
loss_kernel(const float* __restrict__ emb,
            const float* __restrict__ ew,
            const unsigned* __restrict__ msk,
            const float* __restrict__ vtab,
            float* __restrict__ accum)
{
    __shared__ float    As[16 * NDIM];     // 16 KB anchor-row tile
    __shared__ unsigned mrow[16];
    __shared__ float    crow[16];
    __shared__ float    maxred[WAVES][16];
    __shared__ float    redp[WAVES][16];
    __shared__ float    redn[WAVES][16];
    __shared__ float    redc[WAVES][16];

    const int i0    = blockIdx.x * 16;
    const int tid   = threadIdx.x;
    const int wave  = tid >> 5;
    const int lane  = tid & 31;
    const int half  = lane >> 4;
    const int lan16 = lane & 15;

    // stage anchor rows (contiguous) into LDS
    for (int k = tid; k < 16 * NDIM; k += 128)
        As[k] = emb[(size_t)i0 * NDIM + k];
    if (tid < 16) mrow[tid] = msk[i0 + tid];
    __syncthreads();

    const float* Arow = As + lan16 * NDIM;

    // ---------------- pass 1: row max over negatives ----------------
    float rmax[8];
#pragma unroll
    for (int r = 0; r < 8; ++r) rmax[r] = -INFINITY;

    for (int jt = wave; jt < NTILES; jt += WAVES) {
        const int j0 = jt * 16;
        const float* Brow = emb + (size_t)(j0 + lan16) * NDIM;
        v8f acc = sim_tile(Arow, Brow, half);
        const int j = j0 + lan16;
#pragma unroll
        for (int r = 0; r < 8; ++r) {
            const int i = i0 + r + 8 * half;
            float s = acc[r] * INV_TEMP;
            if (j == i) s = -INFINITY;
            rmax[r] = fmaxf(rmax[r], s);
        }
    }
    // reduce across the 16 lanes of each half (offsets stay inside the half)
#pragma unroll
    for (int r = 0; r < 8; ++r) {
        float v = rmax[r];
        v = fmaxf(v, __shfl_xor(v, 1, 32));
        v = fmaxf(v, __shfl_xor(v, 2, 32));
        v = fmaxf(v, __shfl_xor(v, 4, 32));
        v = fmaxf(v, __shfl_xor(v, 8, 32));
        rmax[r] = v;
    }
    if (lan16 == 0) {
#pragma unroll
        for (int r = 0; r < 8; ++r) maxred[wave][r + 8 * half] = rmax[r];
    }
    __syncthreads();
    if (tid < 16) {
        float c = maxred[0][tid];
        for (int w = 1; w < WAVES; ++w) c = fmaxf(c, maxred[w][tid]);
        crow[tid] = c;
    }
    __syncthreads();

    // ---------------- pass 2: stabilized pos/neg sums ----------------
    float cr[8];
    unsigned mr[8];
    float pos[8], neg[8], cnt[8];
#pragma unroll
    for (int r = 0; r < 8; ++r) {
        cr[r] = crow[r + 8 * half];
        mr[r] = mrow[r + 8 * half];
        pos[r] = 0.0f; neg[r] = 0.0f; cnt[r] = 0.0f;
    }

    for (int jt = wave; jt < NTILES; jt += WAVES) {
        const int j0 = jt * 16;
        const float* Brow = emb + (size_t)(j0 + lan16) * NDIM;
        v8f acc = sim_tile(Arow, Brow, half);
        const int j = j0 + lan16;
        const unsigned mj = msk[j];
        const float* vj = vtab + (size_t)j * NCLS;
#pragma unroll
        for (int r = 0; r < 8; ++r) {
            const int i = i0 + r + 8 * half;
            if (j == i) continue;                 // exclude diagonal everywhere
            const float s = acc[r] * INV_TEMP;
            const float e = __expf(s - cr[r]);
            if (mr[r] & mj) {                     // label overlap: positive, weight 1
                pos[r] += e;
                cnt[r] += 1.0f;
                neg[r] += e;
            } else {                              // va weight = pairsim^0.5
                float mx = -INFINITY;
                unsigned mm = mr[r];
                while (mm) {
                    int a = __ffs(mm) - 1;
                    mm &= (mm - 1);
                    mx = fmaxf(mx, vj[a]);
                }
                neg[r] += e * sqrtf(mx);
            }
        }
    }
    // lane reduction (sum) within each 16-lane half
#pragma unroll
    for (int r = 0; r < 8; ++r) {
        float p = pos[r], n = neg[r], c = cnt[r];
        p += __shfl_xor(p, 1, 32); n += __shfl_xor(n, 1, 32); c += __shfl_xor(c, 1, 32);
        p += __shfl_xor(p, 2, 32); n += __shfl_xor(n, 2, 32); c += __shfl_xor(c, 2, 32);
        p += __shfl_xor(p, 4, 32); n += __shfl_xor(n, 4, 32); c += __shfl_xor(c, 4, 32);
        p += __shfl_xor(p, 8, 32); n += __shfl_xor(n, 8, 32); c += __shfl_xor(c, 8, 32);
        pos[r] = p; neg[r] = n; cnt[r] = c;
    }
    if (lan16 == 0) {
#pragma unroll
        for (int r = 0; r < 8; ++r) {
            redp[wave][r + 8 * half] = pos[r];
            redn[wave][r + 8 * half] = neg[r];
            redc[wave][r + 8 * half] = cnt[r];
        }
    }
    __syncthreads();

    if (tid < 16) {
        float p = 0.0f, n = 0.0f, c = 0.0f;
        for (int w = 0; w < WAVES; ++w) {
            p += redp[w][tid];
            n += redn[w][tid];
            c += redc[w][tid];
        }
        if (c > 0.0f) {
            const int i = i0 + tid;
            float loss = -(__logf(p) - __logf(n + 1e-8f)) * ew[i];
            atomicAdd(&accum[0], loss);
            atomicAdd(&accum[1], 1.0f);
        }
    }
}

// ---------------------------------------------------------------------------
// Kernel 3: finalize scalar.
// ---------------------------------------------------------------------------
__global__ void fin_kernel(const float* __restrict__ accum, float* __restrict__ out)
{
    if (threadIdx.x == 0 && blockIdx.x == 0)
        out[0] = accum[0] / fmaxf(accum[1], 1.0f);
}

extern "C" void kernel_launch(void* const* d_in, const int* in_sizes, int n_in,
                              void* d_out, int out_size, void* d_ws, size_t ws_size,
                              hipStream_t stream)
{
    const float* emb    = (const float*)d_in[0];
    const float* logits = (const float*)d_in[1];
    const float* labels = (const float*)d_in[2];
    const float* va     = (const float*)d_in[3];

    float*    ws    = (float*)d_ws;
    float*    ew    = ws;                               // N floats
    unsigned* msk   = (unsigned*)(ws + NROWS);          // N uints
    float*    vtab  = ws + 2 * NROWS;                   // N*C floats
    float*    accum = vtab + (size_t)NROWS * NCLS;      // 2 floats

    prep_kernel<<<NROWS / 256, 256, 0, stream>>>(logits, labels, va, ew, msk, vtab, accum);
    loss_kernel<<<NTILES, 128, 0, stream>>>(emb, ew, msk, vtab, accum);
    fin_kernel<<<1, 32, 0, stream>>>(accum, (float*)d_out);
}